// VectorQuantizer_10127532884670
// MI455X (gfx1250) — compile-verified
//
#include <hip/hip_runtime.h>
#include <math.h>

// ---- problem constants -----------------------------------------------------
#define N_TOT 131072            // 32*64*64 rows
#define DIM   64                // embedding dim
#define KE    1024              // num embeddings
#define RPB   16                // rows per block
#define NBLK  (N_TOT / RPB)     // 8192 main blocks

// ---- d_out flat layout (floats), in reference return order -----------------
// quantized_st [N*D], loss [1], perplexity [1], encodings [N*K],
// encoding_indices [N], distances [N*K]
static const size_t OFF_Q    = 0;
static const size_t OFF_LOSS = 8388608;            // N*D
static const size_t OFF_PERP = 8388609;
static const size_t OFF_ENC  = 8388610;
static const size_t OFF_IDX  = 8388610 + 134217728;        // + N*K
static const size_t OFF_DIST = 8388610 + 134217728 + 131072;

// ---- workspace layout (float units) ----------------------------------------
#define WS_LOSS  0        // [NBLK]  per-block loss partial sums
#define WS_CNT   8192     // [KE]    int32 histogram (codebook usage)
#define WS_ENORM 9216     // [KE]    ||e_k||^2
#define WS_ET    10240    // [KE*DIM] E^T, row-contiguous codebook vectors

typedef float v2f __attribute__((ext_vector_type(2)));
typedef float v8f __attribute__((ext_vector_type(8)));

// ============================================================================
// Kernel 1: e-norms, E^T, zero histogram
// ============================================================================
__global__ __launch_bounds__(256)
void vq_init(const float* __restrict__ E, float* __restrict__ ws) {
    const int k = blockIdx.x * 256 + threadIdx.x;   // 0..1023 (grid = 4 blocks)
    if (k < KE) {
        float s = 0.0f;
        float* et = ws + WS_ET + (size_t)k * DIM;
        #pragma unroll
        for (int d = 0; d < DIM; ++d) {
            float e = E[(size_t)d * KE + k];
            s += e * e;
            et[d] = e;                               // transpose for fast gather
        }
        ws[WS_ENORM + k] = s;
        ((int*)ws)[WS_CNT + k] = 0;
    }
}

// ============================================================================
// Kernel 2: fused distances (WMMA f32) + argmin + all big outputs
// ============================================================================
__global__ __launch_bounds__(256)
void vq_main(const float* __restrict__ X, const float* __restrict__ E,
             float* __restrict__ out, float* __restrict__ ws) {
    __shared__ __align__(16) float sA[RPB * DIM];     // 4 KB  x-tile
    __shared__ __align__(16) float sEn[KE];           // 4 KB  ||e||^2
    __shared__ float sRowN[RPB];                      //       ||x_row||^2
    __shared__ __align__(16) float sDist[RPB * KE];   // 64 KB distance slab
    __shared__ float sMinV[256];
    __shared__ int   sMinI[256];
    __shared__ int   sIdx[RPB];

    const int tid  = threadIdx.x;
    const int row0 = blockIdx.x * RPB;

    // ---- stage x-tile (1024 floats) and e-norms into LDS, coalesced float4
    ((float4*)sA)[tid]  = ((const float4*)(X + (size_t)row0 * DIM))[tid];
    ((float4*)sEn)[tid] = ((const float4*)(ws + WS_ENORM))[tid];
    __syncthreads();

    if (tid < RPB) {
        float s = 0.0f;
        #pragma unroll
        for (int d = 0; d < DIM; ++d) { float v = sA[tid * DIM + d]; s += v * v; }
        sRowN[tid] = s;
    }
    __syncthreads();

    // ---- WMMA phase: each wave owns 128 columns = 8 tiles of 16x16
    const int wave = tid >> 5;
    const int lane = tid & 31;
    const int m16  = lane & 15;
    const int kh   = lane >> 4;          // K half-select for A/B fragments
    const int mbase = kh * 8;            // C/D row group for this lane half

    for (int t = 0; t < 8; ++t) {
        const int n = wave * 128 + t * 16 + m16;   // output column
        v8f acc = {};
        #pragma unroll
        for (int ks = 0; ks < 16; ++ks) {          // K = 64 in steps of 4
            const int k0 = ks * 4 + kh * 2;
            v2f a, b;
            a.x = sA[m16 * DIM + k0];
            a.y = sA[m16 * DIM + k0 + 1];
            b.x = E[(size_t)k0 * KE + n];
            b.y = E[(size_t)(k0 + 1) * KE + n];
            // D = A(16x4,f32) * B(4x16,f32) + C  — exact fp32 matrix op
            acc = __builtin_amdgcn_wmma_f32_16x16x4_f32(
                false, a, false, b, (short)0, acc, false, false);
        }
        #pragma unroll
        for (int e = 0; e < 8; ++e) {
            const int m = mbase + e;
            sDist[m * KE + n] = sRowN[m] - 2.0f * acc[e] + sEn[n];
        }
    }
    __syncthreads();

    // ---- argmin per row (tie-break: lowest index, == argmax(-d))
    {
        const int r   = tid >> 4;                  // row 0..15
        const int seg = tid & 15;                  // 64-entry segment
        const float* rp = sDist + r * KE + seg * 64;
        float bv = 3.402823e38f; int bi = 0;
        for (int j = 0; j < 64; ++j) {
            float v = rp[j];
            if (v < bv) { bv = v; bi = seg * 64 + j; }
        }
        sMinV[tid] = bv; sMinI[tid] = bi;
    }
    __syncthreads();
    if (tid < RPB) {
        float bv = 3.402823e38f; int bi = 0;
        for (int s = 0; s < 16; ++s) {             // ascending: first-min wins
            float v = sMinV[tid * 16 + s];
            if (v < bv) { bv = v; bi = sMinI[tid * 16 + s]; }
        }
        sIdx[tid] = bi;
        atomicAdd(&((int*)ws)[WS_CNT + bi], 1);    // deterministic int histogram
        out[OFF_IDX + (size_t)row0 + tid] = (float)bi;
    }
    __syncthreads();

    // ---- distances + one-hot encodings: coalesced float4 stores
    {
        const float4* sD4 = (const float4*)sDist;
        float4* gd4 = (float4*)(out + OFF_DIST + (size_t)row0 * KE);
        float4* ge4 = (float4*)(out + OFF_ENC  + (size_t)row0 * KE);
        for (int i = tid; i < RPB * KE / 4; i += 256) {
            const int r  = i >> 8;                 // row
            const int kb = (i & 255) * 4;          // first column of the quad
            gd4[i] = sD4[i];
            const int bi = sIdx[r];
            float4 ev;
            ev.x = (kb + 0 == bi) ? 1.0f : 0.0f;
            ev.y = (kb + 1 == bi) ? 1.0f : 0.0f;
            ev.z = (kb + 2 == bi) ? 1.0f : 0.0f;
            ev.w = (kb + 3 == bi) ? 1.0f : 0.0f;
            ge4[i] = ev;
        }
    }

    // ---- quantized (== quantized_st forward) + per-block loss partial
    float lp = 0.0f;
    for (int i = tid; i < RPB * DIM; i += 256) {
        const int r = i >> 6;
        const int d = i & 63;
        const float q = ws[WS_ET + (size_t)sIdx[r] * DIM + d];  // contiguous gather
        out[OFF_Q + (size_t)row0 * DIM + i] = q;
        const float diff = q - sA[i];
        lp += diff * diff;
    }
    sMinV[tid] = lp;
    __syncthreads();
    for (int s = 128; s > 0; s >>= 1) {            // fixed-order tree reduce
        if (tid < s) sMinV[tid] += sMinV[tid + s];
        __syncthreads();
    }
    if (tid == 0) ws[WS_LOSS + blockIdx.x] = sMinV[0];
}

// ============================================================================
// Kernel 3: scalar loss + perplexity (deterministic fixed-order reductions)
// ============================================================================
__global__ __launch_bounds__(256)
void vq_finalize(float* __restrict__ out, const float* __restrict__ ws) {
    __shared__ float red[256];
    const int tid = threadIdx.x;

    float s = 0.0f;
    for (int i = tid; i < NBLK; i += 256) s += ws[WS_LOSS + i];
    red[tid] = s; __syncthreads();
    for (int st = 128; st > 0; st >>= 1) {
        if (tid < st) red[tid] += red[tid + st];
        __syncthreads();
    }
    const float totalSq = red[0];
    __syncthreads();

    const int* cnt = (const int*)ws + WS_CNT;
    float h = 0.0f;
    for (int i = tid; i < KE; i += 256) {
        const float p = (float)cnt[i] * (1.0f / (float)N_TOT);
        h += p * logf(p + 1e-10f);
    }
    red[tid] = h; __syncthreads();
    for (int st = 128; st > 0; st >>= 1) {
        if (tid < st) red[tid] += red[tid + st];
        __syncthreads();
    }
    if (tid == 0) {
        // forward: q_latent == e_latent numerically -> loss = 1.25 * MSE
        out[OFF_LOSS] = 1.25f * totalSq / (float)((size_t)N_TOT * DIM);
        out[OFF_PERP] = expf(-red[0]);
    }
}

// ============================================================================
extern "C" void kernel_launch(void* const* d_in, const int* in_sizes, int n_in,
                              void* d_out, int out_size, void* d_ws, size_t ws_size,
                              hipStream_t stream) {
    const float* X = (const float*)d_in[0];   // x [32,64,64,64] f32
    const float* E = (const float*)d_in[1];   // embeddings [64,1024] f32
    float* out = (float*)d_out;
    float* ws  = (float*)d_ws;

    vq_init    <<<4,    256, 0, stream>>>(E, ws);
    vq_main    <<<NBLK, 256, 0, stream>>>(X, E, out, ws);
    vq_finalize<<<1,    256, 0, stream>>>(out, ws);
}